// GNN_57234734187243
// MI455X (gfx1250) — compile-verified
//
#include <hip/hip_runtime.h>
#include <hip/hip_bf16.h>

typedef __attribute__((ext_vector_type(2))) float v2f;
typedef __attribute__((ext_vector_type(8))) float v8f;

#define LDS_STRIDE 68   // 16x64 f32 tile padded: bank = (4*row + k) % 64, conflict-free A reads

__device__ __forceinline__ void atomAddF(float* p, float v) {
    unsafeAtomicAdd(p, v);   // lowers to global_atomic_add_f32 on CDNA
}

// ---------------- zero workspace ----------------
__global__ __launch_bounds__(256) void GNN_zero_kernel(float* __restrict__ p, int n) {
    int i = blockIdx.x * blockDim.x + threadIdx.x;
    if (i < n) p[i] = 0.0f;
}

// ---------------- edge phase: ea = attr@We+be ; msg = relu(x[src]+ea) ; scatter-add to agg[dst]
__global__ __launch_bounds__(256) void GNN_edge_kernel(
    const float* __restrict__ x, const float* __restrict__ edge_attr,
    const float* __restrict__ We, const float* __restrict__ be,
    const long long* __restrict__ edge_index, float* __restrict__ agg, int n_edges)
{
    int e = blockIdx.x * blockDim.x + threadIdx.x;
    if (e >= n_edges) return;
    const float4 a4 = ((const float4*)edge_attr)[e];       // [E,4], 16B aligned
    const long long src = edge_index[e];
    const long long dst = edge_index[n_edges + e];
    const float* xs = x + src * 12;
    float* ag = agg + dst * 12;
#pragma unroll
    for (int c = 0; c < 12; ++c) {
        float v = be[c] + a4.x * We[c] + a4.y * We[12 + c]
                        + a4.z * We[24 + c] + a4.w * We[36 + c];
        v += xs[c];
        v = v > 0.0f ? v : 0.0f;
        atomAddF(&ag[c], v);
    }
}

// ---------------- node MLP: h = x+agg ; relu(h@W1+b1)@W2+b2 ; relu -> h_out [N,64]
// One wave32 per 16-node tile. f32 WMMA 16x16x4.
// A layout (16x4 f32): lane L<16 holds (M=L, K=0..1), lane L+16 holds (M=L, K=2..3).
// B layout (4x16 f32): lane L<16 holds (N=L, K=0..1), lane L+16 holds (N=L, K=2..3).
// C/D layout: VGPR g -> (M = g + 8*(lane>>4), N = lane&15).
__global__ __launch_bounds__(64) void GNN_node_mlp_wmma(
    const float* __restrict__ x, const float* __restrict__ agg,
    const float* __restrict__ W1, const float* __restrict__ b1,
    const float* __restrict__ W2, const float* __restrict__ b2,
    float* __restrict__ h_out, int n_tiles)
{
    __shared__ float lds[2][16 * LDS_STRIDE];
    const int wave = threadIdx.x >> 5;
    const int lane = threadIdx.x & 31;
    const int tile = blockIdx.x * 2 + wave;
    if (tile >= n_tiles) return;          // uniform per wave: EXEC stays all-ones
    const int node0 = tile * 16;

    const int row  = lane & 15;           // M for A / N for B,C,D
    const int half = lane >> 4;
    const int koff = half * 2;            // K offset inside a 4-slice

    // ---- layer 1: [16x12] @ [12x64] ----
    v8f c1[4];
    {
        v8f z = {};
        c1[0] = z; c1[1] = z; c1[2] = z; c1[3] = z;
    }
#pragma unroll
    for (int kk = 0; kk < 3; ++kk) {
        const int kbase = kk * 4 + koff;
        const int aoff = (node0 + row) * 12 + kbase;
        v2f a;
        a.x = x[aoff]     + agg[aoff];
        a.y = x[aoff + 1] + agg[aoff + 1];
#pragma unroll
        for (int nt = 0; nt < 4; ++nt) {
            const int col = nt * 16 + row;
            v2f b;
            b.x = W1[kbase * 64 + col];
            b.y = W1[(kbase + 1) * 64 + col];
            c1[nt] = __builtin_amdgcn_wmma_f32_16x16x4_f32(
                false, a, false, b, (short)0, c1[nt], false, false);
        }
    }
    // bias + relu -> LDS activation tile (row-major 16x64, padded stride)
    float* act = lds[wave];
#pragma unroll
    for (int nt = 0; nt < 4; ++nt) {
        const int col  = nt * 16 + row;
        const float bb = b1[col];
#pragma unroll
        for (int g = 0; g < 8; ++g) {
            float v = c1[nt][g] + bb;
            v = v > 0.0f ? v : 0.0f;
            act[(g + 8 * half) * LDS_STRIDE + col] = v;   // same-wave LDS: DS ops in-order
        }
    }

    // ---- layer 2: [16x64] @ [64x64] ----
    v8f c2[4];
    {
        v8f z = {};
        c2[0] = z; c2[1] = z; c2[2] = z; c2[3] = z;
    }
#pragma unroll
    for (int kk = 0; kk < 16; ++kk) {
        const int kbase = kk * 4 + koff;
        v2f a;
        a.x = act[row * LDS_STRIDE + kbase];
        a.y = act[row * LDS_STRIDE + kbase + 1];
#pragma unroll
        for (int nt = 0; nt < 4; ++nt) {
            const int col = nt * 16 + row;
            v2f b;
            b.x = W2[kbase * 64 + col];
            b.y = W2[(kbase + 1) * 64 + col];
            c2[nt] = __builtin_amdgcn_wmma_f32_16x16x4_f32(
                false, a, false, b, (short)0, c2[nt], false, false);
        }
    }
    // bias + relu -> h_out
#pragma unroll
    for (int nt = 0; nt < 4; ++nt) {
        const int col  = nt * 16 + row;
        const float bb = b2[col];
#pragma unroll
        for (int g = 0; g < 8; ++g) {
            float v = c2[nt][g] + bb;
            v = v > 0.0f ? v : 0.0f;
            h_out[(long)(node0 + g + 8 * half) * 64 + col] = v;
        }
    }
}

// ---------------- pooling accumulation: sums[batch[n]] += h[n], counts[batch[n]] += 1
__global__ __launch_bounds__(256) void GNN_pool_kernel(
    const float* __restrict__ h, const long long* __restrict__ batch,
    float* __restrict__ sums, float* __restrict__ counts, int n_nodes)
{
    int n = blockIdx.x * blockDim.x + threadIdx.x;
    if (n >= n_nodes) return;
    const long long g = batch[n];
    const float4* hr = (const float4*)(h + (long)n * 64);
    float* s = sums + g * 64;
#pragma unroll
    for (int q = 0; q < 16; ++q) {
        float4 v = hr[q];
        atomAddF(&s[q * 4 + 0], v.x);
        atomAddF(&s[q * 4 + 1], v.y);
        atomAddF(&s[q * 4 + 2], v.z);
        atomAddF(&s[q * 4 + 3], v.w);
    }
    atomAddF(&counts[g], 1.0f);
}

// ---------------- head: mean, relu(pooled@Wh1+bh1)@Wh2+bh2 -> out [G,1]
__global__ __launch_bounds__(256) void GNN_head_kernel(
    const float* __restrict__ sums, const float* __restrict__ counts,
    const float* __restrict__ Wh1, const float* __restrict__ bh1,
    const float* __restrict__ Wh2, const float* __restrict__ bh2,
    float* __restrict__ out, int n_graphs)
{
    int g = blockIdx.x * blockDim.x + threadIdx.x;
    if (g >= n_graphs) return;
    const float inv = 1.0f / fmaxf(counts[g], 1.0f);
    float pooled[64];
#pragma unroll
    for (int c = 0; c < 64; ++c) pooled[c] = sums[g * 64 + c] * inv;
    float acc = bh2[0];
#pragma unroll 4
    for (int j = 0; j < 32; ++j) {
        float hv = bh1[j];
#pragma unroll
        for (int c = 0; c < 64; ++c) hv = fmaf(pooled[c], Wh1[c * 32 + j], hv);
        hv = fmaxf(hv, 0.0f);
        acc = fmaf(hv, Wh2[j], acc);
    }
    out[g] = acc;
}

extern "C" void kernel_launch(void* const* d_in, const int* in_sizes, int n_in,
                              void* d_out, int out_size, void* d_ws, size_t ws_size,
                              hipStream_t stream) {
    const float* x          = (const float*)d_in[0];
    const float* edge_attr  = (const float*)d_in[1];
    const float* We         = (const float*)d_in[2];
    const float* be         = (const float*)d_in[3];
    const float* W1         = (const float*)d_in[4];
    const float* b1         = (const float*)d_in[5];
    const float* W2         = (const float*)d_in[6];
    const float* b2         = (const float*)d_in[7];
    const float* Wh1        = (const float*)d_in[8];
    const float* bh1        = (const float*)d_in[9];
    const float* Wh2        = (const float*)d_in[10];
    const float* bh2        = (const float*)d_in[11];
    const long long* edge_index = (const long long*)d_in[12];
    const long long* batch      = (const long long*)d_in[13];
    float* out = (float*)d_out;

    const int N = in_sizes[0] / 12;   // 100000
    const int E = in_sizes[1] / 4;    // 3200000
    const int G = out_size;           // 2048

    // workspace layout (zeroed region first, then h_out so float4 alignment holds)
    float* agg    = (float*)d_ws;               // [N,12]
    float* sums   = agg + (long)N * 12;         // [G,64]
    float* counts = sums + (long)G * 64;        // [G]
    float* h_out  = counts + G;                 // [N,64]
    const int zeroN = N * 12 + G * 64 + G;

    GNN_zero_kernel<<<(zeroN + 255) / 256, 256, 0, stream>>>(agg, zeroN);

    GNN_edge_kernel<<<(E + 255) / 256, 256, 0, stream>>>(
        x, edge_attr, We, be, edge_index, agg, E);

    const int n_tiles = (N + 15) / 16;          // 6250 (exact)
    GNN_node_mlp_wmma<<<(n_tiles + 1) / 2, 64, 0, stream>>>(
        x, agg, W1, b1, W2, b2, h_out, n_tiles);

    GNN_pool_kernel<<<(N + 255) / 256, 256, 0, stream>>>(
        h_out, batch, sums, counts, N);

    GNN_head_kernel<<<(G + 255) / 256, 256, 0, stream>>>(
        sums, counts, Wh1, bh1, Wh2, bh2, out, G);
}